// DechunkingLayer_89472758710929
// MI455X (gfx1250) — compile-verified
//
#include <hip/hip_runtime.h>
#include <stdint.h>

typedef float v4f __attribute__((ext_vector_type(4)));
typedef int   v4i __attribute__((ext_vector_type(4)));

#define BB  8
#define TT  4096
#define NCC 2048
#define DD  1024
#define D4  (DD / 4)   // 256 float4 per row

// ---------------------------------------------------------------------------
// Phase 1: idx[b,t] = min(exclusive_cumsum(bflags[b,:])[t], NC-1)
// One block per batch row, 256 threads, 16 elements/thread, LDS block scan.
// ---------------------------------------------------------------------------
__global__ __launch_bounds__(256) void dechunk_scan_kernel(const int* __restrict__ bflags,
                                                           int* __restrict__ idx) {
    __shared__ int partial[256];
    const int tid  = threadIdx.x;
    const int row  = blockIdx.x;              // batch index
    const int base = row * TT + tid * 16;     // 16 elements per thread

    const v4i* src = (const v4i*)(bflags + base);
    v4i a0 = src[0], a1 = src[1], a2 = src[2], a3 = src[3];
    int v[16] = {a0.x, a0.y, a0.z, a0.w,  a1.x, a1.y, a1.z, a1.w,
                 a2.x, a2.y, a2.z, a2.w,  a3.x, a3.y, a3.z, a3.w};

    int s = 0;
#pragma unroll
    for (int i = 0; i < 16; ++i) s += v[i];

    partial[tid] = s;
    __syncthreads();
#pragma unroll
    for (int off = 1; off < 256; off <<= 1) {
        int add = (tid >= off) ? partial[tid - off] : 0;
        __syncthreads();
        partial[tid] += add;
        __syncthreads();
    }
    int run = partial[tid] - s;   // exclusive prefix entering this thread's chunk

    int o[16];
#pragma unroll
    for (int i = 0; i < 16; ++i) {
        o[i] = (run < NCC - 1) ? run : (NCC - 1);
        run += v[i];
    }

    v4i* dst = (v4i*)(idx + base);
    v4i r0 = {o[0],  o[1],  o[2],  o[3]};
    v4i r1 = {o[4],  o[5],  o[6],  o[7]};
    v4i r2 = {o[8],  o[9],  o[10], o[11]};
    v4i r3 = {o[12], o[13], o[14], o[15]};
    dst[0] = r0; dst[1] = r1; dst[2] = r2; dst[3] = r3;
}

// ---------------------------------------------------------------------------
// Phase 2: out[b,t,:] = p[b,t]*z[b,idx[t],:] + (1-p[b,t])*z[b,idx[t-1],:]
// (t==0 passes z[b,idx[0],:] through; idx[0]==0).
// One block per (b,t); thread i owns float4 i of the 1024-wide row.
// Gathered z rows are staged with global_load_async_to_lds_b128 (ASYNCcnt):
// each lane stages and consumes only its own 16 bytes, so one per-wave
// s_wait_asynccnt 0 is the only fence needed (no workgroup barrier).
// idx[t]==idx[t-1] for ~half of all rows (b is Bernoulli(0.5)); the indices
// are block-uniform scalars, so a scalar branch skips the duplicate 4 KB
// gather + LDS read in that case. Output leaves through non-temporal b128
// stores so the 134 MB write stream does not evict the 64 MB z working set
// from the 192 MB L2.
// ---------------------------------------------------------------------------
__global__ __launch_bounds__(256) void dechunk_blend_kernel(const v4f* __restrict__ z,
                                                            const float* __restrict__ p,
                                                            const int* __restrict__ idx,
                                                            v4f* __restrict__ out) {
    __shared__ v4f rowA[256];
    __shared__ v4f rowB[256];

    const int tid = threadIdx.x;
    const int bt  = blockIdx.x;          // 0 .. B*T-1
    const int bb  = bt >> 12;            // / 4096
    const int t   = bt & (TT - 1);

    const int   i0 = idx[bt];
    const int   i1 = (t == 0) ? i0 : idx[bt - 1];
    const float pp = (t == 0) ? 1.0f : p[bt];

    const unsigned long long gA =
        (unsigned long long)(uintptr_t)(z + ((size_t)bb * NCC + (size_t)i0) * D4 + tid);
    // Generic LDS pointers carry the wave-relative LDS byte offset in the low
    // 32 bits (ISA 10.2 aperture mapping) — exactly the async VDST operand.
    const uint32_t la = (uint32_t)(uintptr_t)(&rowA[tid]);

    asm volatile("global_load_async_to_lds_b128 %0, %1, off"
                 :: "v"(la), "v"(gA) : "memory");

    v4f a, r;
    if (i1 == i0) {                      // block-uniform: single-gather path
        asm volatile("s_wait_asynccnt 0x0" ::: "memory");
        a = rowA[tid];
        r = a;
    } else {
        const unsigned long long gB =
            (unsigned long long)(uintptr_t)(z + ((size_t)bb * NCC + (size_t)i1) * D4 + tid);
        const uint32_t lb = (uint32_t)(uintptr_t)(&rowB[tid]);
        asm volatile("global_load_async_to_lds_b128 %0, %1, off"
                     :: "v"(lb), "v"(gB) : "memory");
        asm volatile("s_wait_asynccnt 0x0" ::: "memory");
        a = rowA[tid];
        r = rowB[tid];
    }

    const float q = 1.0f - pp;
    v4f o = pp * a + q * r;

    __builtin_nontemporal_store(o, out + (size_t)bt * D4 + tid);
}

// ---------------------------------------------------------------------------
extern "C" void kernel_launch(void* const* d_in, const int* in_sizes, int n_in,
                              void* d_out, int out_size, void* d_ws, size_t ws_size,
                              hipStream_t stream) {
    (void)in_sizes; (void)n_in; (void)out_size; (void)ws_size;

    const float* z      = (const float*)d_in[0];   // [B, NC, D] f32
    const float* p      = (const float*)d_in[1];   // [B, T]     f32
    const int*   bflags = (const int*)  d_in[2];   // [B, T]     i32
    int*         idx    = (int*)d_ws;              // [B, T]     i32 scratch (128 KB)

    dechunk_scan_kernel<<<BB, 256, 0, stream>>>(bflags, idx);
    dechunk_blend_kernel<<<BB * TT, 256, 0, stream>>>((const v4f*)z, p, idx, (v4f*)d_out);
}